// DeformableDETR_27504970563835
// MI455X (gfx1250) — compile-verified
//
#include <hip/hip_runtime.h>
#include <math.h>

// ---------------------------------------------------------------------------
// Deformable-DETR multi-scale deformable attention for MI455X (gfx1250).
// B=2, Lq=Lv=13294, D_MODEL=256, H=8, L=4, P=4, HD=32.
// SPATIAL = (100,100),(50,50),(25,25),(13,13); level starts 0,10000,12500,13125
// ---------------------------------------------------------------------------

typedef float v2f __attribute__((ext_vector_type(2)));
typedef float v4f __attribute__((ext_vector_type(4)));
typedef float v8f __attribute__((ext_vector_type(8)));
typedef unsigned int u32x4 __attribute__((ext_vector_type(4)));
typedef int i32x4 __attribute__((ext_vector_type(4)));
typedef int i32x8 __attribute__((ext_vector_type(8)));

#define D_MODEL 256
#define NHEAD   8
#define HDIM    32
#define NLVL    4
#define NPT     4
#define LQ      13294
#define LV      13294
#define NBATCH  2

// swizzled LDS index for the transposed weight tile:
// element (c, k) of the 64x256 tile lives at  c*256 + (k ^ ((c & 15) << 2))
__device__ __forceinline__ int wswz(int c, int k) {
    return c * 256 + (k ^ ((c & 15) << 2));
}

// ---------------------------------------------------------------------------
// GEMM  C[M,N] = A[M,256] * W[256,N] + bias[N]   (fp32, V_WMMA_F32_16X16X4_F32)
// Block = 256 threads = 8 waves; each wave computes a 16x64 tile.
// W tile (64 cols x 256 k) staged once per block into XOR-swizzled LDS ->
// conflict-free ds_load_b64 B-fragments. Bias tile staged via the Tensor Data
// Mover (tensor_load_to_lds + s_wait_tensorcnt).
// grid.x = N/64, grid.y = ceil(M/128).
// ---------------------------------------------------------------------------
__global__ __launch_bounds__(256) void gemm_k256_wmma_f32(
    const float* __restrict__ A, const float* __restrict__ W,
    const float* __restrict__ bias, float* __restrict__ C,
    int M, int N)
{
    constexpr int K = 256;
    __shared__ float Wt[64 * 256];   // 64 KB swizzled transposed weight tile
    __shared__ float Bs[64];         // bias tile (filled by TDM)

    const int lane = threadIdx.x & 31;
    const int wave = threadIdx.x >> 5;
    const int half = lane >> 4;      // A frag: 0 -> k,k+1 ; 1 -> k+2,k+3
    const int lm   = lane & 15;

    const int m0 = (blockIdx.y * 8 + wave) * 16;
    const int n0 = blockIdx.x * 64;

    // ---- TDM: DMA the 64-entry bias tile into LDS (wave 0 issues) ----------
    if (threadIdx.x < 32) {          // wave-uniform branch; TDM ignores EXEC
        const unsigned lds_base = (unsigned)(uintptr_t)(&Bs[0]);
        const unsigned long long ga =
            (unsigned long long)(const void*)(bias + n0);
        u32x4 g0;
        g0[0] = 1u;                                        // count=1 (valid D#)
        g0[1] = lds_base;                                  // lds_addr (bytes)
        g0[2] = (unsigned)(ga & 0xFFFFFFFFull);            // global_addr lo
        g0[3] = (unsigned)((ga >> 32) & 0x1FFFFFFull)      // global_addr hi
              | (2u << 30);                                // type = 2 (image)
        i32x8 g1;
        g1[0] = (int)(2u << 16);        // workgroup_mask=0, data_size=4B
        g1[1] = (int)(64u << 16);       // tensor_dim0 = 64  (low 16 bits)
        g1[2] = (int)(1u << 16);        // tensor_dim0 hi=0 | tensor_dim1 = 1
        g1[3] = (int)(64u << 16);       // tensor_dim1 hi=0 | tile_dim0 = 64
        g1[4] = 1;                      // tile_dim1 = 1, tile_dim2 = 0
        g1[5] = 64;                     // tensor_dim0_stride = 64
        g1[6] = 0;
        g1[7] = 0;
        i32x4 g2; g2[0] = 0; g2[1] = 0; g2[2] = 0; g2[3] = 0;
        i32x4 g3; g3[0] = 0; g3[1] = 0; g3[2] = 0; g3[3] = 0;
        i32x8 g4; g4[0] = 0; g4[1] = 0; g4[2] = 0; g4[3] = 0;
                  g4[4] = 0; g4[5] = 0; g4[6] = 0; g4[7] = 0;
        // 6-arg toolchain form: groups 0..3, extra group, cache-policy
        __builtin_amdgcn_tensor_load_to_lds(g0, g1, g2, g3, g4, 0);
    }

    // ---- cooperative W-tile staging (overlapped with the TDM transfer) -----
    // thread t: columns c4..c4+3 (b128 global reads), k rows strided by 16
    {
        const int c4 = (threadIdx.x & 15) * 4;
        const int kk = threadIdx.x >> 4;          // 0..15
#pragma unroll 4
        for (int kb = 0; kb < 16; ++kb) {
            const int k = kb * 16 + kk;
            const v4f w4 = *(const v4f*)(W + (size_t)k * N + n0 + c4);
#pragma unroll
            for (int j = 0; j < 4; ++j)
                Wt[wswz(c4 + j, k)] = w4[j];
        }
    }
    if (threadIdx.x < 32) __builtin_amdgcn_s_wait_tensorcnt(0);
    __syncthreads();

    // ---- main K loop --------------------------------------------------------
    int ar = m0 + lm;
    if (ar >= M) ar = M - 1;                      // clamp; stores predicated
    const float* __restrict__ Arow = A + (size_t)ar * K;

    const int xorm = lm << 2;                     // per-lane XOR swizzle term
    const int cb0 = (lm +  0) * 256;
    const int cb1 = (lm + 16) * 256;
    const int cb2 = (lm + 32) * 256;
    const int cb3 = (lm + 48) * 256;

    v8f acc[4];
#pragma unroll
    for (int s = 0; s < 4; ++s)
#pragma unroll
        for (int r = 0; r < 8; ++r) acc[s][r] = 0.f;

#pragma unroll 8
    for (int k = 0; k < K; k += 4) {
        const int ka = k + half * 2;              // A/B fragment K pair
        const int ko = ka ^ xorm;                 // swizzled LDS k offset
        const v2f a  = *(const v2f*)(Arow + ka);          // global b64
        const v2f b0 = *(const v2f*)(&Wt[cb0 + ko]);      // ds_load_b64,
        const v2f b1 = *(const v2f*)(&Wt[cb1 + ko]);      // conflict-free
        const v2f b2 = *(const v2f*)(&Wt[cb2 + ko]);
        const v2f b3 = *(const v2f*)(&Wt[cb3 + ko]);
        acc[0] = __builtin_amdgcn_wmma_f32_16x16x4_f32(false, a, false, b0,
                                                       (short)0, acc[0], false, false);
        acc[1] = __builtin_amdgcn_wmma_f32_16x16x4_f32(false, a, false, b1,
                                                       (short)0, acc[1], false, false);
        acc[2] = __builtin_amdgcn_wmma_f32_16x16x4_f32(false, a, false, b2,
                                                       (short)0, acc[2], false, false);
        acc[3] = __builtin_amdgcn_wmma_f32_16x16x4_f32(false, a, false, b3,
                                                       (short)0, acc[3], false, false);
    }

    // ---- epilogue: D layout lane lm = column, VGPR r = row (r + 8*half) ----
#pragma unroll
    for (int s = 0; s < 4; ++s) {
        const int n  = n0 + s * 16 + lm;
        const float bv = Bs[s * 16 + lm];
#pragma unroll
        for (int r = 0; r < 8; ++r) {
            const int m = m0 + r + 8 * half;
            if (m < M) C[(size_t)m * N + n] = acc[s][r] + bv;
        }
    }
}

// ---------------------------------------------------------------------------
// Sampling stage: one wave per (b, q, h); lane = channel (HD == warpSize == 32).
// Fuses softmax over the 16 attention logits, bilinear sampling with zero
// padding on 4 levels x 4 points, and the weighted reduction.
// Feature maps (13.6 MB/batch) are L2-resident; each tap is a coalesced
// 32-lane x 4B read.
// ---------------------------------------------------------------------------
__device__ __forceinline__ float msda_tap(const float* __restrict__ vl,
                                          int xx, int yy, int w_, int h_,
                                          float wt, int lane)
{
    if ((unsigned)xx < (unsigned)w_ && (unsigned)yy < (unsigned)h_)
        return wt * vl[((size_t)((size_t)yy * w_ + xx)) * D_MODEL + lane];
    return 0.f;
}

__global__ __launch_bounds__(256) void msda_sample_kernel(
    const float* __restrict__ vproj,   // [B, Lv, 256]   (H*HD channel-minor)
    const float* __restrict__ refp,    // [B, Lq, L, 2]
    const float* __restrict__ offs,    // [B, Lq, 256]   (H,L,P,2)
    const float* __restrict__ logits,  // [B, Lq, 128]   (H,16)
    float* __restrict__ outp)          // [B, Lq, 256]
{
    const int lane = threadIdx.x & 31;
    const int wave = threadIdx.x >> 5;
    const long gw  = (long)blockIdx.x * 8 + wave;      // over B*Lq*H
    const long NW  = (long)NBATCH * LQ * NHEAD;
    if (gw >= NW) return;                               // no barriers below

    const int  h  = (int)(gw & 7);
    const long bq = gw >> 3;                           // b*Lq + q
    const long b  = bq / LQ;

    // ---- softmax over the 16 logits of this (b,q,h) (vector b128 loads) ----
    const float* __restrict__ lg = logits + bq * (NHEAD * 16) + h * 16;
    float e[16];
    {
        v4f l4[4];
#pragma unroll
        for (int i = 0; i < 4; ++i) l4[i] = *(const v4f*)(lg + 4 * i);
#pragma unroll
        for (int i = 0; i < 16; ++i) e[i] = l4[i >> 2][i & 3];
    }
    float mx = e[0];
#pragma unroll
    for (int i = 1; i < 16; ++i) mx = fmaxf(mx, e[i]);
    float ssum = 0.f;
#pragma unroll
    for (int i = 0; i < 16; ++i) { e[i] = __expf(e[i] - mx); ssum += e[i]; }
    const float inv = 1.f / ssum;

    // reference points (2x b128) and offsets (8x b128)
    v4f rp2[2];
    {
        const float* __restrict__ rp = refp + bq * (NLVL * 2);
        rp2[0] = *(const v4f*)(rp);
        rp2[1] = *(const v4f*)(rp + 4);
    }
    v4f of4[8];
    {
        const float* __restrict__ of = offs + bq * D_MODEL + h * (NLVL * NPT * 2);
#pragma unroll
        for (int i = 0; i < 8; ++i) of4[i] = *(const v4f*)(of + 4 * i);
    }

    const float* __restrict__ vb =
        vproj + (size_t)b * LV * D_MODEL + h * HDIM;

    const int hh[NLVL] = {100, 50, 25, 13};
    const int ww[NLVL] = {100, 50, 25, 13};
    const int st[NLVL] = {0, 10000, 12500, 13125};

    float acc = 0.f;
#pragma unroll
    for (int l = 0; l < NLVL; ++l) {
        const float rx = rp2[l >> 1][(l & 1) * 2 + 0];
        const float ry = rp2[l >> 1][(l & 1) * 2 + 1];
        const float wf = (float)ww[l];
        const float hf = (float)hh[l];
        const float* __restrict__ vl = vb + (size_t)st[l] * D_MODEL;
#pragma unroll
        for (int p = 0; p < NPT; ++p) {
            const int pt = l * NPT + p;                  // 0..15
            const float ox = of4[pt >> 1][(pt & 1) * 2 + 0];
            const float oy = of4[pt >> 1][(pt & 1) * 2 + 1];
            // grid = 2*loc - 1 ; sample coord = loc*dim - 0.5 (align_corners=F)
            const float x = (rx + ox / wf) * wf - 0.5f;
            const float y = (ry + oy / hf) * hf - 0.5f;
            const float x0f = floorf(x), y0f = floorf(y);
            const int   x0 = (int)x0f,  y0 = (int)y0f;
            const float wx1 = x - x0f, wy1 = y - y0f;
            const float wx0 = 1.f - wx1, wy0 = 1.f - wy1;
            const float aw = e[pt] * inv;
            float s;
            s  = msda_tap(vl, x0,     y0,     ww[l], hh[l], wx0 * wy0, lane);
            s += msda_tap(vl, x0 + 1, y0,     ww[l], hh[l], wx1 * wy0, lane);
            s += msda_tap(vl, x0,     y0 + 1, ww[l], hh[l], wx0 * wy1, lane);
            s += msda_tap(vl, x0 + 1, y0 + 1, ww[l], hh[l], wx1 * wy1, lane);
            acc += aw * s;
        }
    }
    outp[bq * D_MODEL + h * HDIM + lane] = acc;
}

// ---------------------------------------------------------------------------
// Host-side orchestration (graph-capture safe: only kernel launches on stream)
// ---------------------------------------------------------------------------
extern "C" void kernel_launch(void* const* d_in, const int* in_sizes, int n_in,
                              void* d_out, int out_size, void* d_ws, size_t ws_size,
                              hipStream_t stream)
{
    const float* query  = (const float*)d_in[0];
    const float* refp   = (const float*)d_in[1];
    const float* value  = (const float*)d_in[2];
    /* d_in[3] = spatial_shapes (compile-time constants here) */
    const float* W_off  = (const float*)d_in[4];
    const float* b_off  = (const float*)d_in[5];
    const float* W_attn = (const float*)d_in[6];
    const float* b_attn = (const float*)d_in[7];
    const float* W_v    = (const float*)d_in[8];
    const float* b_v    = (const float*)d_in[9];
    const float* W_out  = (const float*)d_in[10];
    const float* b_out  = (const float*)d_in[11];
    float* out = (float*)d_out;

    const long MQ = (long)NBATCH * LQ;     // 26588 rows for every GEMM

    float* ws    = (float*)d_ws;
    float* vproj = ws;                     // [MQ, 256]
    float* offb  = vproj + MQ * 256;       // [MQ, 256]
    float* logit = offb  + MQ * 256;       // [MQ, 128]
    float* outp  = logit + MQ * 128;       // [MQ, 256]

    const dim3 blk(256);
    const dim3 g256(256 / 64, (unsigned)((MQ + 127) / 128));
    const dim3 g128(128 / 64, (unsigned)((MQ + 127) / 128));

    // 1) value projection
    gemm_k256_wmma_f32<<<g256, blk, 0, stream>>>(value, W_v, b_v, vproj,
                                                 (int)MQ, 256);
    // 2) sampling offsets
    gemm_k256_wmma_f32<<<g256, blk, 0, stream>>>(query, W_off, b_off, offb,
                                                 (int)MQ, 256);
    // 3) attention logits
    gemm_k256_wmma_f32<<<g128, blk, 0, stream>>>(query, W_attn, b_attn, logit,
                                                 (int)MQ, 128);
    // 4) fused softmax + bilinear sampling + weighted reduce
    {
        const long NW = (long)NBATCH * LQ * NHEAD;     // 212704 waves
        const unsigned nblk = (unsigned)((NW + 7) / 8);
        msda_sample_kernel<<<nblk, blk, 0, stream>>>(vproj, refp, offb, logit,
                                                     outp);
    }
    // 5) output projection
    gemm_k256_wmma_f32<<<g256, blk, 0, stream>>>(outp, W_out, b_out, out,
                                                 (int)MQ, 256);
}